// GConv_6322191859838
// MI455X (gfx1250) — compile-verified
//
#include <hip/hip_runtime.h>

#define N_NODES  100000
#define N_EDGES  1600000
#define DIM      128
#define N_LAYERS 3
#define N_GRAPHS 128
#define OUTD     (N_LAYERS * DIM)   // 384
#define BN_EPS   1e-5f

typedef __attribute__((ext_vector_type(2))) float v2f;
typedef __attribute__((ext_vector_type(8))) float v8f;

#if defined(__gfx1250__) && __has_builtin(__builtin_amdgcn_wmma_f32_16x16x4_f32)
#define HAVE_WMMA_F32 1
#endif

__device__ __forceinline__ void atomic_add_f32(float* p, float v) {
    __hip_atomic_fetch_add(p, v, __ATOMIC_RELAXED, __HIP_MEMORY_SCOPE_AGENT);
}

// ---------------------------------------------------------------- utilities
__global__ void fill_zero(float* __restrict__ p, int n) {
    int i = blockIdx.x * blockDim.x + threadIdx.x;
    if (i < n) p[i] = 0.0f;
}

// aggr[node][:] = z[node][:]  (GIN self term; zstride = 128 for x, 384 inside z_cat)
__global__ void init_aggr(const float* __restrict__ z, int zstride,
                          float* __restrict__ aggr) {
    int t    = blockIdx.x * 256 + threadIdx.x;   // N_NODES*32 threads exactly
    int node = t >> 5;
    int c4   = (t & 31) << 2;
    float4 v = *(const float4*)(z + (size_t)node * zstride + c4);
    *(float4*)(aggr + (size_t)node * DIM + c4) = v;
}

// aggr[dst][:] += z[src][:]   (32 lanes per edge, float4 gather, 4 f32 atomics)
__global__ void edge_scatter(const float* __restrict__ z, int zstride,
                             const int* __restrict__ src, const int* __restrict__ dst,
                             float* __restrict__ aggr) {
    int t  = blockIdx.x * 256 + threadIdx.x;     // N_EDGES*32 threads exactly
    int e  = t >> 5;
    int c4 = (t & 31) << 2;
    int s  = src[e];
    int d  = dst[e];
    float4 v = *(const float4*)(z + (size_t)s * zstride + c4);
    float* o = aggr + (size_t)d * DIM + c4;
    atomic_add_f32(o + 0, v.x);
    atomic_add_f32(o + 1, v.y);
    atomic_add_f32(o + 2, v.z);
    atomic_add_f32(o + 3, v.w);
}

// ---------------------------------------------------------- fused GIN MLP
// hout = relu( relu(hin @ W1 + b1) @ W2 + b2 ),  hin = z + aggr (already summed)
// Block: 256 threads = 8 waves; block owns 16 rows; wave w owns cols [16w,16w+16).
// fp32 WMMA 16x16x4, K swept in steps of 4 (32 MACs per 16x16 tile per GEMM).
__global__ __launch_bounds__(256) void gin_mlp(
        const float* __restrict__ hin,
        const float* __restrict__ W1, const float* __restrict__ b1,
        const float* __restrict__ W2, const float* __restrict__ b2,
        float* __restrict__ hout) {
    __shared__ float As[16][DIM + 4];   // input tile (pad -> conflict-free col reads)
    __shared__ float Hs[16][DIM + 4];   // hidden tile (never leaves LDS)
    __shared__ float bs1[DIM], bs2[DIM];

    const int tid  = threadIdx.x;
    const int row0 = blockIdx.x * 16;

    if (tid < DIM) { bs1[tid] = b1[tid]; bs2[tid] = b2[tid]; }
    for (int i = tid; i < 16 * DIM; i += 256)
        As[i >> 7][i & 127] = hin[(size_t)(row0 + (i >> 7)) * DIM + (i & 127)];
    __syncthreads();

    const int lane = tid & 31;
    const int wave = tid >> 5;
    const int lo   = lane & 15;       // row (A) / col (B,C,D) within tile
    const int hi   = lane >> 4;       // half-wave selector
    const int c    = wave * 16 + lo;  // output column in [0,128)

    // ---- GEMM1: Hs = relu(As @ W1 + b1) ----
    v8f acc = {};
#ifdef HAVE_WMMA_F32
    for (int k = 0; k < DIM; k += 4) {
        v2f a, b;
        // A 16x4 layout: lanes 0-15 -> K=k,k+1 ; lanes 16-31 -> K=k+2,k+3
        a.x = As[lo][k + 2 * hi];
        a.y = As[lo][k + 2 * hi + 1];
        // B 4x16 layout mirrors A's K striping across half-waves
        b.x = W1[(size_t)(k + 2 * hi) * DIM + c];
        b.y = W1[(size_t)(k + 2 * hi + 1) * DIM + c];
        acc = __builtin_amdgcn_wmma_f32_16x16x4_f32(false, a, false, b,
                                                    (short)0, acc, false, false);
    }
#else
    for (int k = 0; k < DIM; ++k) {
        float wk = W1[(size_t)k * DIM + c];
        #pragma unroll
        for (int i = 0; i < 8; ++i) acc[i] += As[i + 8 * hi][k] * wk;
    }
#endif
    {
        float bb = bs1[c];
        #pragma unroll
        for (int i = 0; i < 8; ++i) {
            float v = acc[i] + bb;                 // D: vgpr i -> row i (+8 for hi half)
            Hs[i + 8 * hi][c] = v > 0.f ? v : 0.f;
        }
    }
    __syncthreads();

    // ---- GEMM2: hout = relu(Hs @ W2 + b2) ----
    v8f acc2 = {};
#ifdef HAVE_WMMA_F32
    for (int k = 0; k < DIM; k += 4) {
        v2f a, b;
        a.x = Hs[lo][k + 2 * hi];
        a.y = Hs[lo][k + 2 * hi + 1];
        b.x = W2[(size_t)(k + 2 * hi) * DIM + c];
        b.y = W2[(size_t)(k + 2 * hi + 1) * DIM + c];
        acc2 = __builtin_amdgcn_wmma_f32_16x16x4_f32(false, a, false, b,
                                                     (short)0, acc2, false, false);
    }
#else
    for (int k = 0; k < DIM; ++k) {
        float wk = W2[(size_t)k * DIM + c];
        #pragma unroll
        for (int i = 0; i < 8; ++i) acc2[i] += Hs[i + 8 * hi][k] * wk;
    }
#endif
    {
        float bb = bs2[c];
        #pragma unroll
        for (int i = 0; i < 8; ++i) {
            float v = acc2[i] + bb;
            hout[(size_t)(row0 + i + 8 * hi) * DIM + c] = v > 0.f ? v : 0.f;
        }
    }
}

// ------------------------------------------------------------- BatchNorm
// stats layout: [0,128) sum  [128,256) sumsq  [256,384) mean  [384,512) rstd
__global__ void bn_stats(const float* __restrict__ h, float* __restrict__ stats) {
    int col = threadIdx.x;   // blockDim = 128
    float s = 0.f, ss = 0.f;
    for (int r = blockIdx.x; r < N_NODES; r += gridDim.x) {
        float v = h[(size_t)r * DIM + col];
        s += v; ss += v * v;
    }
    atomic_add_f32(&stats[col], s);
    atomic_add_f32(&stats[DIM + col], ss);
}

__global__ void bn_finalize(float* __restrict__ stats) {
    int c = threadIdx.x;     // blockDim = 128
    float inv_n = 1.0f / (float)N_NODES;
    float mean  = stats[c] * inv_n;
    float var   = stats[DIM + c] * inv_n - mean * mean;
    stats[2 * DIM + c] = mean;
    stats[3 * DIM + c] = rsqrtf(var + BN_EPS);
}

// normalize -> write z_cat slice -> atomic sum-pool into g_cat
__global__ void bn_apply_pool(const float* __restrict__ h, const float* __restrict__ stats,
                              const float* __restrict__ gamma, const float* __restrict__ beta,
                              const int* __restrict__ batch,
                              float* __restrict__ zcat, float* __restrict__ gcat, int l) {
    int t    = blockIdx.x * 256 + threadIdx.x;   // N_NODES*32 threads exactly
    int node = t >> 5;
    int c4   = (t & 31) << 2;
    int g    = batch[node];
    float4 v = *(const float4*)(h + (size_t)node * DIM + c4);
    const float* mean = stats + 2 * DIM;
    const float* rstd = stats + 3 * DIM;
    float4 o;
    o.x = (v.x - mean[c4 + 0]) * rstd[c4 + 0] * gamma[c4 + 0] + beta[c4 + 0];
    o.y = (v.y - mean[c4 + 1]) * rstd[c4 + 1] * gamma[c4 + 1] + beta[c4 + 1];
    o.z = (v.z - mean[c4 + 2]) * rstd[c4 + 2] * gamma[c4 + 2] + beta[c4 + 2];
    o.w = (v.w - mean[c4 + 3]) * rstd[c4 + 3] * gamma[c4 + 3] + beta[c4 + 3];
    *(float4*)(zcat + (size_t)node * OUTD + l * DIM + c4) = o;
    float* gp = gcat + (size_t)g * OUTD + l * DIM + c4;
    atomic_add_f32(gp + 0, o.x);
    atomic_add_f32(gp + 1, o.y);
    atomic_add_f32(gp + 2, o.z);
    atomic_add_f32(gp + 3, o.w);
}

// ---------------------------------------------------------------- driver
extern "C" void kernel_launch(void* const* d_in, const int* in_sizes, int n_in,
                              void* d_out, int out_size, void* d_ws, size_t ws_size,
                              hipStream_t stream) {
    const float* x     = (const float*)d_in[0];
    const int*   ei    = (const int*)d_in[1];
    const int*   src   = ei;
    const int*   dst   = ei + N_EDGES;
    const int*   batch = (const int*)d_in[2];
    const float* W1    = (const float*)d_in[3];
    const float* b1    = (const float*)d_in[4];
    const float* W2    = (const float*)d_in[5];
    const float* b2    = (const float*)d_in[6];
    const float* gamma = (const float*)d_in[7];
    const float* beta  = (const float*)d_in[8];

    float* zcat = (float*)d_out;                         // [N_NODES, 384]
    float* gcat = zcat + (size_t)N_NODES * OUTD;         // [N_GRAPHS, 384]

    float* aggr  = (float*)d_ws;                         // [N_NODES, 128]
    float* h     = aggr + (size_t)N_NODES * DIM;         // [N_NODES, 128]
    float* stats = h + (size_t)N_NODES * DIM;            // [4*128]

    // pooled output accumulated via atomics -> must start at zero
    fill_zero<<<(N_GRAPHS * OUTD + 255) / 256, 256, 0, stream>>>(gcat, N_GRAPHS * OUTD);

    const float* z = x;
    int zstride = DIM;
    for (int l = 0; l < N_LAYERS; ++l) {
        init_aggr<<<N_NODES * 32 / 256, 256, 0, stream>>>(z, zstride, aggr);
        edge_scatter<<<N_EDGES * 32 / 256, 256, 0, stream>>>(z, zstride, src, dst, aggr);
        gin_mlp<<<N_NODES / 16, 256, 0, stream>>>(aggr,
                                                  W1 + (size_t)l * DIM * DIM, b1 + (size_t)l * DIM,
                                                  W2 + (size_t)l * DIM * DIM, b2 + (size_t)l * DIM,
                                                  h);
        fill_zero<<<2, 256, 0, stream>>>(stats, 4 * DIM);
        bn_stats<<<2048, DIM, 0, stream>>>(h, stats);
        bn_finalize<<<1, DIM, 0, stream>>>(stats);
        bn_apply_pool<<<N_NODES * 32 / 256, 256, 0, stream>>>(h, stats,
                                                              gamma + (size_t)l * DIM,
                                                              beta + (size_t)l * DIM,
                                                              batch, zcat, gcat, l);
        // next layer consumes this layer's normalized output in-place from z_cat
        z = zcat + (size_t)l * DIM;
        zstride = OUTD;
    }
}